// GCNN_loss_58110907515297
// MI455X (gfx1250) — compile-verified
//
#include <hip/hip_runtime.h>
#include <math.h>
#include <stdint.h>

// Problem constants from the reference: B=8, T=2048, D=512, fp32 nodes, int64 labels.
#define NB 8
#define NT 2048
#define ND 512
#define EPSV 1e-8f
// LDS row stride in floats: 512 + 2 pad dwords after each 256 stored dwords
// (matches TDM pad_interval=256dw / pad_amount=2dw). Element (r,c) lives at
// r*516 + c + 2*(c>=256). Fragment reads then hit banks (4n+k+2h)%64 -> no conflicts.
#define LSTRIDE 516

typedef __attribute__((ext_vector_type(2))) float v2f;
typedef __attribute__((ext_vector_type(8))) float v8f;
typedef __attribute__((ext_vector_type(4))) unsigned int u32x4;
typedef __attribute__((ext_vector_type(8))) int i32x8;
typedef __attribute__((ext_vector_type(4))) int i32x4;

#if __has_builtin(__builtin_amdgcn_tensor_load_to_lds)
#define USE_TDM 1
#else
#define USE_TDM 0
#endif

#if USE_TDM
// Issue one TDM 2-D tile load: tile_rows x 512 f32 from a 2048x512 tensor into
// LDS at lds_off, with hardware padding producing the 516-float row stride.
// This toolchain (clang-23 / therock-10.0) exposes the 6-arg builtin:
//   (u32x4 group0, i32x8 group1, i32x4 group2, i32x4 group3, i32x8, i32 cpol)
__device__ __forceinline__ void tdm_load_2d(unsigned lds_off, const float* gaddr,
                                            int tile_rows) {
  const unsigned long long ga = (unsigned long long)(uintptr_t)gaddr;
  u32x4 g0;
  g0[0] = 1u;                                              // count=1, user D#
  g0[1] = lds_off;                                         // lds_addr (bytes)
  g0[2] = (unsigned)(ga & 0xFFFFFFFFu);                    // global_addr[31:0]
  g0[3] = (unsigned)((ga >> 32) & 0x1FFFFFFu) | (2u << 30);// addr[56:32] | type=2
  i32x8 g1;
  g1[0] = (2 << 16)      // data_size = 4 bytes
        | (1 << 20)      // pad_enable
        | (7 << 22)      // pad_interval code 7 = 256 dwords
        | (1 << 25);     // pad_amount  code 1 = 2 dwords
  g1[1] = (ND & 0xFFFF) << 16;   // tensor_dim0[15:0] at bits[63:48]
  g1[2] = (NT & 0xFFFF) << 16;   // tensor_dim1[15:0] at bits[95:80]
  g1[3] = (ND & 0xFFFF) << 16;   // tile_dim0 at bits[127:112]
  g1[4] = tile_rows & 0xFFFF;    // tile_dim1 at [143:128]; tile_dim2 = 0
  g1[5] = ND;                    // tensor_dim0_stride[31:0]
  g1[6] = 0;                     // stride0 hi / dim1_stride lo
  g1[7] = 0;
  const i32x4 gz4 = {0, 0, 0, 0};               // 2-D tile: groups 2/3 unused
  const i32x8 gz8 = {0, 0, 0, 0, 0, 0, 0, 0};
  __builtin_amdgcn_tensor_load_to_lds(g0, g1, gz4, gz4, gz8, 0);
}
#endif

// Manual padded-layout fill (fallback, and structure-compatible with TDM layout).
__device__ __forceinline__ void fill_tile(float* dstLds, const float* src,
                                          int rows, int tid, int nthreads) {
  for (int idx = tid; idx < rows * (ND / 4); idx += nthreads) {
    const int r = idx >> 7, c = (idx & 127) * 4;           // ND/4 = 128 float4
    float4 v = *(const float4*)(src + (size_t)r * ND + c);
    *(float4*)(dstLds + r * LSTRIDE + c + ((c >= 256) ? 2 : 0)) = v;
  }
}

// ---------------- Kernel 1: per-row squared norms ----------------
__global__ __launch_bounds__(256) void row_sqnorm_kernel(
    const float* __restrict__ x, float* __restrict__ sq) {
  const int wave = (blockIdx.x * blockDim.x + threadIdx.x) >> 5;
  const int lane = threadIdx.x & 31;
  if (wave >= NB * NT) return;
  const float* row = x + (size_t)wave * ND;
  float s = 0.f;
#pragma unroll
  for (int i = 0; i < ND / 128; ++i) {
    float4 v = *(const float4*)(row + (lane + 32 * i) * 4);
    s += v.x * v.x + v.y * v.y + v.z * v.z + v.w * v.w;
  }
#pragma unroll
  for (int off = 16; off >= 1; off >>= 1) s += __shfl_xor(s, off, 32);
  if (lane == 0) sq[wave] = s;
}

// ---------------- Kernel 2: fused Gram (WMMA f32) + hard mining ----------------
// 128 threads = 4 wave32s; each wave owns 16 rows (block: 64 rows of one batch).
// A strip staged once by one TDM; B column tiles double-buffered via TDM so the
// DMA of tile jt+1 overlaps the 128 V_WMMA_F32_16X16X4_F32 of tile jt.
__global__ __launch_bounds__(128, 1) void hardmine_kernel(
    const float* __restrict__ x, const long long* __restrict__ lab,
    const float* __restrict__ sq, float* __restrict__ loss) {
  __shared__ float As[64 * LSTRIDE];        // 132 KB
  __shared__ float Bs[2 * 16 * LSTRIDE];    //  66 KB (double buffer)

  const int b = blockIdx.y;
  const int rowblock = blockIdx.x * 64;
  const size_t bT = (size_t)b * NT;
  const float* xb = x + bT * ND;
  const int t = threadIdx.x;
  const int lane = t & 31;
  const int wv = t >> 5;        // wave id 0..3
  const int n = lane & 15;      // fragment M/N lane index
  const int h = lane >> 4;      // lane half -> K sub-offset 2h

  // Stage A strip (64 rows) + first B tile.
#if USE_TDM
  if (wv == 0) {
    const unsigned as_off = (unsigned)(uintptr_t)(void*)As;
    const unsigned bs_off = (unsigned)(uintptr_t)(void*)Bs;
    tdm_load_2d(as_off, xb + (size_t)rowblock * ND, 64);
    tdm_load_2d(bs_off, xb, 16);
    __builtin_amdgcn_s_wait_tensorcnt(0);
  }
#else
  fill_tile(As, xb + (size_t)rowblock * ND, 64, t, 128);
  fill_tile(Bs, xb, 16, t, 128);
#endif
  __syncthreads();

  const int rowbase = rowblock + wv * 16;
  float sqr[8];
  int pbr[8];
#pragma unroll
  for (int e = 0; e < 8; ++e) {             // accum rows M=e (h=0) / M=8+e (h=1)
    const int row = rowbase + e + 8 * h;
    sqr[e] = sq[bT + row];
    pbr[e] = (int)(lab[bT + row] != 0);     // pos_branch per reference
  }

  float bmaxd2[8], bmaxdot[8], bmind2[8], bmindot[8];
  int bmaxi[8], bmini[8];
#pragma unroll
  for (int e = 0; e < 8; ++e) {
    bmaxd2[e] = -__builtin_inff(); bmaxdot[e] = 0.f; bmaxi[e] = -1;
    bmind2[e] =  __builtin_inff(); bmindot[e] = 0.f; bmini[e] = -1;
  }

  const float* Ap = As + (wv * 16 + n) * LSTRIDE + 2 * h;

  for (int jt = 0; jt < NT / 16; ++jt) {
#if USE_TDM
    if (wv == 0 && jt + 1 < NT / 16) {      // kick DMA for next tile (other buffer)
      const unsigned bs_off = (unsigned)(uintptr_t)(void*)Bs;
      tdm_load_2d(bs_off + (unsigned)(((jt + 1) & 1) * (16 * LSTRIDE * 4)),
                  xb + (size_t)(jt + 1) * 16 * ND, 16);
    }
#endif
    const float* Bp = Bs + (jt & 1) * (16 * LSTRIDE) + n * LSTRIDE + 2 * h;

    const int col = jt * 16 + n;
    const float sqc = sq[bT + col];
    const int labc = (int)lab[bT + col];

    // 16x16 dot tile via 128x V_WMMA_F32_16X16X4_F32.
    // A lane l: M=l&15, K=k+2h+{0,1}; B lane l: N=l&15, same K -> same addressing.
    v8f c = {};
#pragma unroll 8
    for (int k = 0; k < 256; k += 4) {
      v2f a  = *(const v2f*)(Ap + k);
      v2f bb = *(const v2f*)(Bp + k);
      c = __builtin_amdgcn_wmma_f32_16x16x4_f32(false, a, false, bb,
                                                (short)0, c, false, false);
    }
#pragma unroll 8
    for (int k = 256; k < 512; k += 4) {    // +2: hardware pad after 256 dwords
      v2f a  = *(const v2f*)(Ap + k + 2);
      v2f bb = *(const v2f*)(Bp + k + 2);
      c = __builtin_amdgcn_wmma_f32_16x16x4_f32(false, a, false, bb,
                                                (short)0, c, false, false);
    }

    const bool is2 = (labc == 2), is0 = (labc == 0);
#pragma unroll
    for (int e = 0; e < 8; ++e) {
      const float dot = c[e];
      const float d2 = sqr[e] + sqc - 2.0f * dot;
      const bool same = pbr[e] ? is2 : is0;
      const bool opp  = pbr[e] ? is0 : is2;
      if (same && d2 > bmaxd2[e]) { bmaxd2[e] = d2; bmaxdot[e] = dot; bmaxi[e] = col; }
      if (opp  && d2 < bmind2[e]) { bmind2[e] = d2; bmindot[e] = dot; bmini[e] = col; }
    }

#if USE_TDM
    if (wv == 0) __builtin_amdgcn_s_wait_tensorcnt(0);  // next tile landed
#else
    if (jt + 1 < NT / 16) {
      __syncthreads();  // all reads of the other buffer (iter jt-1) are done
      fill_tile(Bs + ((jt + 1) & 1) * (16 * LSTRIDE),
                xb + (size_t)(jt + 1) * 16 * ND, 16, t, 128);
    }
#endif
    __syncthreads();
  }

  // Per-row reduction across the 16 lanes of each half; value ties break on
  // smaller column index => first occurrence, like jnp.arg{max,min}.
#pragma unroll
  for (int e = 0; e < 8; ++e) {
    float vd = bmaxd2[e], vt = bmaxdot[e]; int vi = bmaxi[e];
    float md = bmind2[e], mt = bmindot[e]; int mi = bmini[e];
#pragma unroll
    for (int off = 8; off >= 1; off >>= 1) {
      float od = __shfl_xor(vd, off, 16);
      float ot = __shfl_xor(vt, off, 16);
      int   oi = __shfl_xor(vi, off, 16);
      if (od > vd || (od == vd && (unsigned)oi < (unsigned)vi)) { vd = od; vt = ot; vi = oi; }
      float pd = __shfl_xor(md, off, 16);
      float pt = __shfl_xor(mt, off, 16);
      int   pi = __shfl_xor(mi, off, 16);
      if (pd < md || (pd == md && (unsigned)pi < (unsigned)mi)) { md = pd; mt = pt; mi = pi; }
    }
    if (n == 0) {
      const int row = rowbase + e + 8 * h;
      const float ni = fmaxf(sqrtf(sqr[e]), EPSV);
      float l = 0.f;
      if (vi >= 0) {  // has_same: 1 - cos(x_i, x_far)
        const float nf = fmaxf(sqrtf(sq[bT + vi]), EPSV);
        l += 1.0f - vt / (ni * nf);
      }
      if (mi >= 0) {  // has_opp: + cos(x_i, x_near)
        const float nn = fmaxf(sqrtf(sq[bT + mi]), EPSV);
        l += mt / (ni * nn);
      }
      loss[bT + row] = l;
    }
  }
}

// ---------------- Kernel 3: deterministic final reduction ----------------
__global__ __launch_bounds__(256) void reduce_kernel(
    const float* __restrict__ loss, float* __restrict__ out) {
  __shared__ float sm[256];
  float s = 0.f;
  for (int i = threadIdx.x; i < NB * NT; i += 256) s += loss[i];
  sm[threadIdx.x] = s;
  __syncthreads();
  for (int off = 128; off >= 1; off >>= 1) {
    if ((int)threadIdx.x < off) sm[threadIdx.x] += sm[threadIdx.x + off];
    __syncthreads();
  }
  if (threadIdx.x == 0) out[0] = sm[0] / (float)(NB * NT);   // GCNN_WEIGHT = 1
}

extern "C" void kernel_launch(void* const* d_in, const int* in_sizes, int n_in,
                              void* d_out, int out_size, void* d_ws, size_t ws_size,
                              hipStream_t stream) {
  (void)in_sizes; (void)n_in; (void)out_size; (void)ws_size;
  const float* nodes = (const float*)d_in[0];
  const long long* labels = (const long long*)d_in[1];  // int64 labels
  float* out = (float*)d_out;
  float* sq   = (float*)d_ws;            // NB*NT floats (64 KB)
  float* loss = sq + NB * NT;            // NB*NT floats (64 KB)

  // 1) row norms: 16384 rows, 8 waves/block -> 2048 blocks
  row_sqnorm_kernel<<<(NB * NT) / 8, 256, 0, stream>>>(nodes, sq);

  // 2) fused WMMA Gram + hard mining: 32 row-blocks x 8 batches = 256 blocks
  dim3 grid(NT / 64, NB);
  hardmine_kernel<<<grid, 128, 0, stream>>>(nodes, labels, sq, loss);

  // 3) scalar loss
  reduce_kernel<<<1, 256, 0, stream>>>(loss, out);
}